// SumAggregator_22935125360844
// MI455X (gfx1250) — compile-verified
//
#include <hip/hip_runtime.h>

typedef __attribute__((ext_vector_type(2))) float v2f;
typedef __attribute__((ext_vector_type(4))) float v4f;
typedef __attribute__((ext_vector_type(8))) float v8f;

#define BDIM  128   // 4 waves (wave32)
#define ROWS  16    // (b,e) rows per block
#define NNB   32    // neighbors
#define DDIM  64    // feature dim
#define LSTR  66    // padded LDS row stride

__global__ __launch_bounds__(BDIM) void kgnn_fused(
    const float* __restrict__ selfv,   // [BE, 64]
    const float* __restrict__ nbrv,    // [BE, 32, 64]
    const float* __restrict__ relv,    // [BE, 32, 64]
    const float* __restrict__ userv,   // [B, 64]
    const float* __restrict__ W,       // [64, 64]
    const float* __restrict__ blin,    // [64]
    float* __restrict__ out)           // [BE, 64]
{
    __shared__ float Xs[ROWS * LSTR];  // self + aggregated neighbors, 16x64

    const int t    = threadIdx.x & 31;  // lane
    const int wv   = threadIdx.x >> 5;  // wave 0..3
    const int r0   = blockIdx.x * ROWS; // base (b*E+e) row
    const int bidx = r0 >> 5;           // batch index (E=32, 16 | 32)
    const int tl   = t & 15;
    const int h    = t >> 4;

    // Per-lane 4-float slice of the user embedding: dims (4t)&63 .. +3
    // (reused across rows -> keep regular-temporal, stays in WGP$/L2)
    const v4f uv = *reinterpret_cast<const v4f*>(userv + bidx * DDIM + ((4 * t) & 63));

    // ---------------- Phase 1: attention + aggregation (4 rows per wave) --------------
    for (int rr = 0; rr < 4; ++rr) {
        const int  lr  = wv * 4 + rr;
        const long row = (long)(r0 + lr);
        const float* relp = relv + row * (NNB * DDIM);
        const float* nbrp = nbrv + row * (NNB * DDIM);

        // scores: coalesced NT float4 stream; flat idx 128i+4t -> neighbor n=2i+h, dims (4t)&63
        float s = 0.0f;
        #pragma unroll
        for (int i = 0; i < 16; ++i) {
            v4f v = __builtin_nontemporal_load(
                        reinterpret_cast<const v4f*>(relp + 128 * i + 4 * t));
            float p = v.x * uv.x + v.y * uv.y + v.z * uv.z + v.w * uv.w;
            p += __shfl_xor(p, 1);
            p += __shfl_xor(p, 2);
            p += __shfl_xor(p, 4);
            p += __shfl_xor(p, 8);       // half-wave reduce: score of neighbor 2i+h
            if (tl == i) s = p;          // lane t keeps score of neighbor n = 2*tl + h
        }
        s *= (1.0f / (float)DDIM);       // mean over dim

        // softmax over the 32 neighbors (one score per lane)
        float mx = s;
        mx = fmaxf(mx, __shfl_xor(mx, 16));
        mx = fmaxf(mx, __shfl_xor(mx, 8));
        mx = fmaxf(mx, __shfl_xor(mx, 4));
        mx = fmaxf(mx, __shfl_xor(mx, 2));
        mx = fmaxf(mx, __shfl_xor(mx, 1));
        float ev  = __expf(s - mx);
        float sum = ev;
        sum += __shfl_xor(sum, 16);
        sum += __shfl_xor(sum, 8);
        sum += __shfl_xor(sum, 4);
        sum += __shfl_xor(sum, 2);
        sum += __shfl_xor(sum, 1);
        const float w = ev / (sum * (float)NNB);  // attn / N (mean over neighbors)

        // weighted aggregation: same coalesced NT stream over neighbor_vectors
        float ax = 0.f, ay = 0.f, az = 0.f, aw = 0.f;
        #pragma unroll
        for (int i = 0; i < 16; ++i) {
            v4f v = __builtin_nontemporal_load(
                        reinterpret_cast<const v4f*>(nbrp + 128 * i + 4 * t));
            // weight of neighbor n=2i+h lives in lane i + (h<<4)
            float wn = __shfl(w, i + (t & 16));
            ax += wn * v.x; ay += wn * v.y; az += wn * v.z; aw += wn * v.w;
        }
        ax += __shfl_xor(ax, 16);   // combine even/odd neighbor parities (same dims)
        ay += __shfl_xor(ay, 16);
        az += __shfl_xor(az, 16);
        aw += __shfl_xor(aw, 16);

        if (t < 16) {               // lanes 0..15 hold dims 4t..4t+3
            v4f sv = __builtin_nontemporal_load(
                         reinterpret_cast<const v4f*>(selfv + row * DDIM + 4 * t));
            Xs[lr * LSTR + 4 * t + 0] = sv.x + ax;
            Xs[lr * LSTR + 4 * t + 1] = sv.y + ay;
            Xs[lr * LSTR + 4 * t + 2] = sv.z + az;
            Xs[lr * LSTR + 4 * t + 3] = sv.w + aw;
        }
    }

    __syncthreads();

    // ------------- Phase 2: out = relu(X @ W^T + b) via V_WMMA_F32_16X16X4_F32 --------
    // Wave wv computes the 16x16 tile: rows r0..r0+15, cols j0..j0+15.
    const int j0 = wv * 16;
    v8f c = {};                                   // f32 16x16 accumulator (8 VGPRs)
    #pragma unroll
    for (int kk = 0; kk < 16; ++kk) {
        const int k0 = 4 * kk + 2 * h;            // this lane's K pair within the 16x4 tile
        v2f a, b;
        a.x = Xs[tl * LSTR + k0];                 // A[M=tl, K=k0], A[M=tl, K=k0+1]
        a.y = Xs[tl * LSTR + k0 + 1];
        const float* wp = W + (j0 + tl) * DDIM + k0;  // B[K, N=j0+tl] = W[j0+tl, K]
        b.x = wp[0];                              // W reused by all blocks -> regular temporal
        b.y = wp[1];
        c = __builtin_amdgcn_wmma_f32_16x16x4_f32(
                /*neg_a=*/false, a, /*neg_b=*/false, b,
                /*c_mod=*/(short)0, c, /*reuse_a=*/false, /*reuse_b=*/false);
    }

    const float bv = blin[j0 + tl];
    #pragma unroll
    for (int r = 0; r < 8; ++r) {
        const int m = r + 8 * h;                  // C layout: VGPR r -> row r (+8 for hi half)
        float val = c[r] + bv;
        val = fmaxf(val, 0.0f);
        // output written once, never re-read -> NT store (skip write-allocate pollution)
        __builtin_nontemporal_store(val, out + (long)(r0 + m) * DDIM + j0 + tl);
    }
}

extern "C" void kernel_launch(void* const* d_in, const int* in_sizes, int n_in,
                              void* d_out, int out_size, void* d_ws, size_t ws_size,
                              hipStream_t stream) {
    const float* selfv = (const float*)d_in[0];  // self_vectors   [B,E,D]
    const float* nbrv  = (const float*)d_in[1];  // neighbor_vectors [B,E,N,D]
    const float* relv  = (const float*)d_in[2];  // neighbor_relations [B,E,N,D]
    const float* userv = (const float*)d_in[3];  // user_embeddings [B,D]
    // d_in[4] = masks (unused by the module)
    const float* W     = (const float*)d_in[5];  // [D,D]
    const float* blin  = (const float*)d_in[6];  // [D]
    float* out = (float*)d_out;

    const int BE = in_sizes[0] / DDIM;           // B*E rows
    dim3 grid(BE / ROWS), block(BDIM);
    hipLaunchKernelGGL(kgnn_fused, grid, block, 0, stream,
                       selfv, nbrv, relv, userv, W, blin, out);
}